// HawkesRGCNLayer_19696720020159
// MI455X (gfx1250) — compile-verified
//
#include <hip/hip_runtime.h>
#include <hip/hip_bf16.h>

#define D64 64
#define LEAKY 0.01f

typedef __attribute__((ext_vector_type(2))) float v2f;
typedef __attribute__((ext_vector_type(8))) float v8f;

// ---------- helpers: order-preserving float<->uint map for atomic max ----------
__device__ __forceinline__ unsigned f2ord(float f) {
    unsigned u = __float_as_uint(f);
    return (u & 0x80000000u) ? ~u : (u | 0x80000000u);
}
__device__ __forceinline__ float ord2f(unsigned u) {
    unsigned v = (u & 0x80000000u) ? (u & 0x7FFFFFFFu) : ~u;
    return __uint_as_float(v);
}
#define ORD_NEG_INF 0x007FFFFFu   // f2ord(-inf)

// ---------- 0) init: zero d_out (agg accumulator), den; set maxbuf=-inf ----------
__global__ __launch_bounds__(256) void k_init(float* __restrict__ out,
                                              float* __restrict__ den,
                                              unsigned* __restrict__ maxbuf,
                                              int nNodes) {
    int i = blockIdx.x * blockDim.x + threadIdx.x;   // over nNodes*32 float2
    if (i < nNodes * 32) {
        ((float2*)out)[i] = make_float2(0.f, 0.f);
    }
    if (i < nNodes) {
        den[i] = 0.f;
        maxbuf[i] = ORD_NEG_INF;
    }
}

// ---------- 1) node GEMMs via V_WMMA_F32_16X16X4_F32 ----------
// One wave computes a 16-row x 64-col tile.
// A (16x4 f32): lanes 0-15 rows 0-15 K={0,1}; lanes 16-31 rows 0-15 K={2,3}
// B (4x16 f32): lanes 0-15 K={0,1} col n; lanes 16-31 K={2,3} col n
// C (16x16):    vgpr r -> lanes 0-15 (M=r, N=lane), lanes 16-31 (M=r+8, N=lane-16)
__device__ __forceinline__ void gemm_tile16x64(
    const v2f afrag[16], const float* __restrict__ W, float* __restrict__ O,
    int khalf, int m, int rowBase, int nNodes, bool full)
{
#pragma unroll
    for (int nt = 0; nt < 4; ++nt) {
        int col = nt * 16 + m;
        v8f c = {};
#pragma unroll
        for (int k = 0; k < 16; ++k) {
            v2f b;
            b.x = W[(k * 4 + khalf + 0) * D64 + col];
            b.y = W[(k * 4 + khalf + 1) * D64 + col];
            c = __builtin_amdgcn_wmma_f32_16x16x4_f32(
                    false, afrag[k], false, b, (short)0, c, false, false);
        }
        float* op = O + (size_t)rowBase * D64 + col;
        if (full) {
#pragma unroll
            for (int r = 0; r < 8; ++r) op[(size_t)r * D64] = c[r];
        } else {
#pragma unroll
            for (int r = 0; r < 8; ++r)
                if (rowBase + r < nNodes) op[(size_t)r * D64] = c[r];
        }
    }
}

__global__ __launch_bounds__(256) void k_node_gemm3(
    const float* __restrict__ h,
    const float* __restrict__ wn,   // weight_neighbor rows [0:64)
    const float* __restrict__ wl,   // loop_weight
    const float* __restrict__ we,   // evolve_loop_weight
    float* __restrict__ hW, float* __restrict__ hL, float* __restrict__ hE,
    int nNodes)
{
    int wave = (blockIdx.x * blockDim.x + threadIdx.x) >> 5;
    int lane = threadIdx.x & 31;
    int row0 = wave * 16;
    if (row0 >= nNodes) return;

    bool full  = (row0 + 16 <= nNodes);
    int  m     = lane & 15;
    int  khalf = (lane >> 4) ? 2 : 0;

    int arow = row0 + m;
    if (arow > nNodes - 1) arow = nNodes - 1;          // clamp for partial tile
    const float* ap = h + (size_t)arow * D64;

    v2f afrag[16];
#pragma unroll
    for (int k = 0; k < 16; ++k)
        afrag[k] = *(const v2f*)(ap + k * 4 + khalf);  // b64 load, 8B aligned

    int rowBase = row0 + ((lane >> 4) ? 8 : 0);
    gemm_tile16x64(afrag, wn, hW, khalf, m, rowBase, nNodes, full);
    gemm_tile16x64(afrag, wl, hL, khalf, m, rowBase, nNodes, full);
    gemm_tile16x64(afrag, we, hE, khalf, m, rowBase, nNodes, full);
}

// ---------- 2) per-node attention scores: wave-per-node shuffle reduction ----------
__global__ __launch_bounds__(256) void k_node_scores(
    const float* __restrict__ h, const float* __restrict__ attn,
    float* __restrict__ s_src, float* __restrict__ s_dst, int nNodes)
{
    int wave = (blockIdx.x * blockDim.x + threadIdx.x) >> 5;
    int lane = threadIdx.x & 31;
    if (wave >= nNodes) return;
    float2 hv = ((const float2*)(h + (size_t)wave * D64))[lane];
    float2 a1 = ((const float2*)attn)[lane];
    float2 a2 = ((const float2*)(attn + D64))[lane];
    float p1 = hv.x * a1.x + hv.y * a1.y;
    float p2 = hv.x * a2.x + hv.y * a2.y;
#pragma unroll
    for (int off = 16; off > 0; off >>= 1) {
        p1 += __shfl_down(p1, off, 32);
        p2 += __shfl_down(p2, off, 32);
    }
    if (lane == 0) { s_src[wave] = p1; s_dst[wave] = p2; }
}

// ---------- 3) relation projections + scores: wave-per-relation ----------
__global__ __launch_bounds__(256) void k_rel(
    const float* __restrict__ rel_emb, const float* __restrict__ wn_rel, // rows [64:128)
    const float* __restrict__ a3, float* __restrict__ relW, float* __restrict__ s_rel,
    int nRels)
{
    int wave = (blockIdx.x * blockDim.x + threadIdx.x) >> 5;
    int lane = threadIdx.x & 31;
    if (wave >= nRels) return;
    const float* re = rel_emb + (size_t)wave * D64;
    float acc0 = 0.f, acc1 = 0.f;
    int c0 = lane * 2;
    for (int k = 0; k < D64; ++k) {
        float r = re[k];
        acc0 += r * wn_rel[k * D64 + c0];
        acc1 += r * wn_rel[k * D64 + c0 + 1];
    }
    relW[(size_t)wave * D64 + c0]     = acc0;
    relW[(size_t)wave * D64 + c0 + 1] = acc1;
    float2 rv = ((const float2*)re)[lane];
    float2 av = ((const float2*)a3)[lane];
    float p = rv.x * av.x + rv.y * av.y;
#pragma unroll
    for (int off = 16; off > 0; off >>= 1) p += __shfl_down(p, off, 32);
    if (lane == 0) s_rel[wave] = p;
}

// ---------- 4) edge pass A: scores + segment max ----------
__global__ __launch_bounds__(256) void k_edge_scores(
    const int* __restrict__ src, const int* __restrict__ dst, const int* __restrict__ etype,
    const float* __restrict__ etime, const float* __restrict__ delta,
    const float* __restrict__ s_src, const float* __restrict__ s_dst, const float* __restrict__ s_rel,
    float* __restrict__ scores, unsigned* __restrict__ maxbuf, int nEdges)
{
    int e = blockIdx.x * blockDim.x + threadIdx.x;
    if (e >= nEdges) return;
    float t = s_src[src[e]] + s_dst[dst[e]] + s_rel[etype[e]];
    float lr = (t > 0.f) ? t : LEAKY * t;
    float sc = -etime[e] * delta[0] * lr;
    scores[e] = sc;
    atomicMax(&maxbuf[dst[e]], f2ord(sc));
}

// ---------- 5) edge pass B: softmax denominator ----------
__global__ __launch_bounds__(256) void k_edge_den(
    const int* __restrict__ dst, const float* __restrict__ scores,
    const unsigned* __restrict__ maxbuf, float* __restrict__ den, int nEdges)
{
    int e = blockIdx.x * blockDim.x + threadIdx.x;
    if (e >= nEdges) return;
    int d = dst[e];
    float m = ord2f(maxbuf[d]);
    atomicAdd(&den[d], __expf(scores[e] - m));
}

// ---------- 6) edge pass C (heavy): grid-stride wave-per-edge scatter, with
//             prefetch of the next edge's gather row to hide HBM latency ----------
__global__ __launch_bounds__(256) void k_edge_agg(
    const int* __restrict__ src, const int* __restrict__ dst, const int* __restrict__ etype,
    const float* __restrict__ scores, const unsigned* __restrict__ maxbuf,
    const float* __restrict__ den, const float* __restrict__ hW,
    const float* __restrict__ relW, float* __restrict__ agg, int nEdges)
{
    int wave   = (blockIdx.x * blockDim.x + threadIdx.x) >> 5;
    int lane   = threadIdx.x & 31;
    int nWaves = (gridDim.x * blockDim.x) >> 5;

    for (int e = wave; e < nEdges; e += nWaves) {
        int s = src[e], d = dst[e], r = etype[e];

        // prefetch next iteration's random gather row (global_prefetch_b8)
        int en = e + nWaves;
        if (en < nEdges) {
            int sn = src[en];
            __builtin_prefetch(hW + (size_t)sn * D64 + lane * 2, 0, 1);
        }

        float m = ord2f(maxbuf[d]);
        float w = __expf(scores[e] - m) / den[d];
        float2 hw = ((const float2*)(hW   + (size_t)s * D64))[lane];
        float2 rw = ((const float2*)(relW + (size_t)r * D64))[lane];
        float* ap = agg + (size_t)d * D64 + lane * 2;
        atomicAdd(ap,     (hw.x + rw.x) * w);
        atomicAdd(ap + 1, (hw.y + rw.y) * w);
    }
}

// ---------- 7) finalize: out = (has_in ? agg : h) + (has_in ? hL : hE) ----------
__global__ __launch_bounds__(256) void k_finalize(
    const float* __restrict__ h, const float* __restrict__ hL, const float* __restrict__ hE,
    const float* __restrict__ den, float* __restrict__ out, int nNodes)
{
    int i = blockIdx.x * blockDim.x + threadIdx.x;   // over nNodes*32 float2
    if (i >= nNodes * 32) return;
    int n = i >> 5;
    bool has = den[n] > 0.f;
    float2 a  = ((const float2*)out)[i];   // agg accumulated in-place in d_out
    float2 hv = ((const float2*)h)[i];
    float2 lp = has ? ((const float2*)hL)[i] : ((const float2*)hE)[i];
    float2 o;
    o.x = (has ? a.x : hv.x) + lp.x;
    o.y = (has ? a.y : hv.y) + lp.y;
    ((float2*)out)[i] = o;
}

static inline int cdiv(long long a, long long b) { return (int)((a + b - 1) / b); }

extern "C" void kernel_launch(void* const* d_in, const int* in_sizes, int n_in,
                              void* d_out, int out_size, void* d_ws, size_t ws_size,
                              hipStream_t stream)
{
    const float* h       = (const float*)d_in[0];
    const float* rel_emb = (const float*)d_in[1];
    const float* wn      = (const float*)d_in[2];   // (128,64)
    const float* attn    = (const float*)d_in[3];   // 192
    const float* delta   = (const float*)d_in[4];   // 1
    const float* wl      = (const float*)d_in[5];   // (64,64)
    const float* we      = (const float*)d_in[6];   // (64,64)
    const float* etime   = (const float*)d_in[7];
    const int*   src     = (const int*)d_in[8];
    const int*   dst     = (const int*)d_in[9];
    const int*   etype   = (const int*)d_in[10];
    float* out = (float*)d_out;

    const int N = in_sizes[0] / D64;     // 100000
    const int R = in_sizes[1] / D64;     // 200
    const int E = in_sizes[7];           // 1600000

    // workspace layout
    char* p = (char*)d_ws;
    float* hW     = (float*)p; p += (size_t)N * D64 * sizeof(float);
    float* hL     = (float*)p; p += (size_t)N * D64 * sizeof(float);
    float* hE     = (float*)p; p += (size_t)N * D64 * sizeof(float);
    float* scores = (float*)p; p += (size_t)E * sizeof(float);
    float* s_src  = (float*)p; p += (size_t)N * sizeof(float);
    float* s_dst  = (float*)p; p += (size_t)N * sizeof(float);
    float* den    = (float*)p; p += (size_t)N * sizeof(float);
    unsigned* maxbuf = (unsigned*)p; p += (size_t)N * sizeof(unsigned);
    float* relW   = (float*)p; p += (size_t)R * D64 * sizeof(float);
    float* s_rel  = (float*)p; p += (size_t)R * sizeof(float);

    const int B = 256;

    // 0) init (zero agg accumulator in d_out, den; maxbuf = -inf)
    k_init<<<cdiv((long long)N * 32, B), B, 0, stream>>>(out, den, maxbuf, N);

    // 1) node GEMMs (WMMA f32)
    {
        int waves = cdiv(N, 16);
        k_node_gemm3<<<cdiv((long long)waves * 32, B), B, 0, stream>>>(
            h, wn, wl, we, hW, hL, hE, N);
    }

    // 2) per-node attention scores
    k_node_scores<<<cdiv((long long)N * 32, B), B, 0, stream>>>(h, attn, s_src, s_dst, N);

    // 3) relation projection + scores
    k_rel<<<cdiv((long long)R * 32, B), B, 0, stream>>>(
        rel_emb, wn + (size_t)D64 * D64, attn + 2 * D64, relW, s_rel, R);

    // 4) edge scores + segment max
    k_edge_scores<<<cdiv(E, B), B, 0, stream>>>(
        src, dst, etype, etime, delta, s_src, s_dst, s_rel, scores, maxbuf, E);

    // 5) softmax denominators
    k_edge_den<<<cdiv(E, B), B, 0, stream>>>(dst, scores, maxbuf, den, E);

    // 6) weighted message aggregation (grid-stride wave per edge, prefetched)
    {
        int blocks = 6144;                      // persistent waves; ~33 edges each
        long long need = cdiv((long long)E * 32, B);
        if (need < blocks) blocks = (int)need;
        k_edge_agg<<<blocks, B, 0, stream>>>(
            src, dst, etype, scores, maxbuf, den, hW, relW, out, E);
    }

    // 7) finalize
    k_finalize<<<cdiv((long long)N * 32, B), B, 0, stream>>>(h, hL, hE, den, out, N);
}